// PlanMapBoundLoss_14465449853368
// MI455X (gfx1250) — compile-verified
//
#include <hip/hip_runtime.h>

typedef __attribute__((ext_vector_type(2))) float v2f;
typedef __attribute__((ext_vector_type(8))) float v8f;

#define Bn 1024
#define Tn 6
#define Vn 100
#define Pn 20
#define NPT (Vn * Pn)   // 2000
#define NPAD 2048       // padded to multiple of 16*NWAVES
#define NTHREADS 128
#define NWAVES 4
#define NTILES (NPAD / 16)          // 128
#define TILES_PER_WAVE (NTILES / NWAVES)  // 32

__global__ void zero_out_kernel(float* out) { out[0] = 0.0f; }

__launch_bounds__(NTHREADS)
__global__ void plan_map_bound_loss_kernel(const float* __restrict__ ego,
                                           const float* __restrict__ lanes,
                                           const float* __restrict__ scores,
                                           float* __restrict__ out) {
    __shared__ float qx[NPAD];
    __shared__ float qy[NPAD];
    __shared__ float qsq[NPAD];
    __shared__ float predx[8], predy[8], startx[8], starty[8];
    __shared__ int   vmask[Vn];
    __shared__ float wmin[NWAVES][Tn];
    __shared__ int   wminv[NWAVES][Tn];
    __shared__ float tloss[Tn];
    __shared__ int   tinter[Tn];

    const int b   = blockIdx.x;
    const int tid = threadIdx.x;

    // --- lane score mask: not_bound = score[...,2] < 0.5 ---
    if (tid < Vn) {
        vmask[tid] = (scores[(b * Vn + tid) * 3 + 2] < 0.5f) ? 1 : 0;
    }
    // --- cumsum of ego deltas -> pred; starts = shifted pred ---
    if (tid == NTHREADS - 1) {
        float cx = 0.0f, cy = 0.0f;
        for (int t = 0; t < Tn; ++t) {
            startx[t] = cx; starty[t] = cy;
            cx += ego[(b * Tn + t) * 2 + 0];
            cy += ego[(b * Tn + t) * 2 + 1];
            predx[t] = cx; predy[t] = cy;
        }
    }
    __syncthreads();

    // --- stage transformed (and masked) lane points into LDS ---
    for (int j = tid; j < NPAD; j += NTHREADS) {
        float x, y;
        if (j < NPT) {
            int v = j / Pn;
            float rx = lanes[(size_t)(b * NPT + j) * 2 + 0];
            float ry = lanes[(size_t)(b * NPT + j) * 2 + 1];
            x = rx * 30.0f - 15.0f;   // scale = PC_RANGE[3]-PC_RANGE[0], offset
            y = ry * 60.0f - 30.0f;
            if (vmask[v]) { x = 1000000.0f; y = 1000000.0f; }  // MASK_VAL
        } else {
            x = 1.0e7f; y = 1.0e7f;   // padding: never the min
        }
        qx[j] = x; qy[j] = y; qsq[j] = x * x + y * y;
    }
    __syncthreads();

    const int  wave = tid >> 5;
    const int  lane = tid & 31;
    const int  tA   = lane & 15;      // row (M) / col (N) index within half-wave
    const bool hi   = lane >= 16;     // holds K=2,3 for A ; K=2,3 for B

    // A row t: [-2*px, -2*py, px^2+py^2, 1]  (zeros for t >= Tn)
    float px = (tA < Tn) ? predx[tA] : 0.0f;
    float py = (tA < Tn) ? predy[tA] : 0.0f;
    v2f A;
    A.x = (tA < Tn) ? (hi ? (px * px + py * py) : (-2.0f * px)) : 0.0f;
    A.y = (tA < Tn) ? (hi ? 1.0f : (-2.0f * py)) : 0.0f;

    float msq[Tn];
    int   mv[Tn];
#pragma unroll
    for (int r = 0; r < Tn; ++r) { msq[r] = 3.0e38f; mv[r] = 0; }

    // --- WMMA distance tiles: D[t][j] = ||p_t - q_j||^2 ---
    for (int n = 0; n < TILES_PER_WAVE; ++n) {
        int tile = wave + NWAVES * n;
        int jB   = tile * 16 + tA;
        float q0 = qx[jB];
        float q1 = qy[jB];
        float qs = qsq[jB];
        // B col j: [qx, qy, 1, qx^2+qy^2]
        v2f Bm;
        Bm.x = hi ? 1.0f : q0;
        Bm.y = hi ? qs   : q1;
        v8f c = {};
        c = __builtin_amdgcn_wmma_f32_16x16x4_f32(
                false, A, false, Bm, (short)0, c, false, false);
        int vcur = jB / Pn;
#pragma unroll
        for (int r = 0; r < Tn; ++r) {   // valid rows M=0..5 live in lanes 0..15
            float d   = c[r];
            bool  upd = d < msq[r];
            msq[r] = upd ? d    : msq[r];
            mv[r]  = upd ? vcur : mv[r];
        }
    }

    // --- min/argmin across the 16 N-lanes (stays within half-wave) ---
#pragma unroll
    for (int off = 8; off >= 1; off >>= 1) {
#pragma unroll
        for (int r = 0; r < Tn; ++r) {
            float osq = __shfl_xor(msq[r], off, 32);
            int   ov  = __shfl_xor(mv[r],  off, 32);
            bool  upd = osq < msq[r];
            msq[r] = upd ? osq : msq[r];
            mv[r]  = upd ? ov  : mv[r];
        }
    }
    if (lane == 0) {
#pragma unroll
        for (int r = 0; r < Tn; ++r) { wmin[wave][r] = msq[r]; wminv[wave][r] = mv[r]; }
    }
    __syncthreads();

    // --- per-timestep: final min, hinge loss, segment-intersection test ---
    if (tid < Tn) {
        const int t = tid;
        float best = 3.0e38f; int bv = 0;
        for (int w = 0; w < NWAVES; ++w) {
            float s = wmin[w][t];
            if (s < best) { best = s; bv = wminv[w][t]; }
        }
        float mind = sqrtf(best);
        float loss = (mind > 1.0f) ? 0.0f : (1.0f - mind);

        float s1x = startx[t], s1y = starty[t];
        float e1x = predx[t],  e1y = predy[t];
        float d1x = e1x - s1x, d1y = e1y - s1y;
        int hit = 0;
        int base = bv * Pn;
        for (int p = 0; p < Pn - 1; ++p) {
            float s2x = qx[base + p],     s2y = qy[base + p];
            float e2x = qx[base + p + 1], e2y = qy[base + p + 1];
            float d2x = e2x - s2x, d2y = e2y - s2y;
            float det = d1x * d2y - d1y * d2x;
            if (det != 0.0f) {
                float dx = s2x - s1x, dy = s2y - s1y;
                float tt = (dx * d2y - dy * d2x) / det;
                float uu = (dx * d1y - dy * d1x) / det;
                if (tt >= 0.0f && tt <= 1.0f && uu >= 0.0f && uu <= 1.0f) hit = 1;
            }
        }
        tloss[t]  = loss;
        tinter[t] = hit;
    }
    __syncthreads();

    // --- first_t masking + block contribution to the mean ---
    if (tid == 0) {
        int first_t = Tn;
        for (int t = 0; t < Tn; ++t) { if (tinter[t]) { first_t = t; break; } }
        float s = 0.0f;
        for (int t = 0; t < first_t; ++t) s += tloss[t];
        atomicAdd(out, s * (1.0f / (float)(Bn * Tn)));
    }
}

extern "C" void kernel_launch(void* const* d_in, const int* in_sizes, int n_in,
                              void* d_out, int out_size, void* d_ws, size_t ws_size,
                              hipStream_t stream) {
    const float* ego    = (const float*)d_in[0];  // (B, T, 2)
    const float* lanes  = (const float*)d_in[1];  // (B, V, P, 2)
    const float* scores = (const float*)d_in[2];  // (B, V, C)
    float* out = (float*)d_out;                   // scalar

    zero_out_kernel<<<1, 1, 0, stream>>>(out);
    plan_map_bound_loss_kernel<<<Bn, NTHREADS, 0, stream>>>(ego, lanes, scores, out);
}